// la_layer_19189913879292
// MI455X (gfx1250) — compile-verified
//
#include <hip/hip_runtime.h>

typedef __attribute__((ext_vector_type(2))) float v2f;
typedef __attribute__((ext_vector_type(8))) float v8f;

#define BB 8
#define TT 8
#define CC 512
#define WW 56
#define HWN 3136   // 56*56
#define GG 8

// ---------------- Kernel 1: global average pool (first 411MB streaming pass) ----
__global__ void gap_kernel(const float* __restrict__ x,
                           const float* __restrict__ ctx,
                           float* __restrict__ q_mean,   // [B*C]
                           float* __restrict__ k_mean)   // [B*T*C]
{
    const int wid  = blockIdx.x * (blockDim.x >> 5) + (threadIdx.x >> 5);
    const int lane = threadIdx.x & 31;
    const float* src;
    float* dst;
    if (wid < BB * CC) {
        src = x + (size_t)wid * HWN;
        dst = q_mean + wid;
    } else {
        const int r = wid - BB * CC;
        src = ctx + (size_t)r * HWN;
        dst = k_mean + r;
    }
    const float4* s4 = (const float4*)src;
    float sum = 0.f;
    for (int j = lane; j < HWN / 4; j += 32) {
        const float4 v = s4[j];
        sum += v.x + v.y + v.z + v.w;
    }
    #pragma unroll
    for (int off = 16; off > 0; off >>= 1)
        sum += __shfl_down(sum, off, 32);
    if (lane == 0) *dst = sum * (1.0f / (float)HWN);
}

// ---------------- Kernel 2: conv1d + masked-WMMA attention + softmax -----------
// One wave per batch. attn[g,t] = sum_ch A[g][ch]*B[ch][t] with
//   A[m][ch] = Q[ch] iff (ch>>6)==m  (block-diagonal head mask)
//   B[ch][n] = Kconv[n][ch] for n<8
// computed as 128 x v_wmma_f32_16x16x4_f32 over ch=0..511.
__global__ void attn_kernel(const float* __restrict__ q_mean,
                            const float* __restrict__ k_mean,
                            const float* __restrict__ wq,
                            const float* __restrict__ wk,
                            float* __restrict__ attn)   // [B*G*T]
{
    __shared__ float Qs[CC];
    __shared__ float Kc[TT * CC];
    const int b    = blockIdx.x;
    const int lane = threadIdx.x;   // 0..31

    // conv1d 'same' (cross-correlation), k=5, pad=2, zero boundary
    for (int i = lane; i < CC; i += 32) {
        float acc = 0.f;
        #pragma unroll
        for (int j = 0; j < 5; ++j) {
            const int idx = i + j - 2;
            const float v = (idx >= 0 && idx < CC) ? q_mean[b * CC + idx] : 0.f;
            acc += v * wq[j];
        }
        Qs[i] = acc;
    }
    for (int i = lane; i < TT * CC; i += 32) {
        const int t = i >> 9, cix = i & (CC - 1);
        float acc = 0.f;
        #pragma unroll
        for (int j = 0; j < 5; ++j) {
            const int idx = cix + j - 2;
            const float v = (idx >= 0 && idx < CC) ? k_mean[(b * TT + t) * CC + idx] : 0.f;
            acc += v * wk[j];
        }
        Kc[i] = acc;
    }
    __syncthreads();

    // WMMA fragment roles per ISA layout tables (f32 16x16x4)
    const int m     = lane & 15;          // A row / B col / D col index
    const int khalf = (lane >> 4) << 1;   // lane-half K offset (0 or 2)
    const int kr    = m & 7;              // clamped t index for safe LDS reads
    v8f acc = {};
    for (int kk = 0; kk < CC; kk += 4) {
        const int ch0 = kk + khalf;
        const int ch1 = ch0 + 1;
        v2f a, bfrag;
        const float a0 = Qs[ch0], a1 = Qs[ch1];
        a.x = ((ch0 >> 6) == m) ? a0 : 0.f;
        a.y = ((ch1 >> 6) == m) ? a1 : 0.f;
        const float b0 = Kc[kr * CC + ch0], b1 = Kc[kr * CC + ch1];
        bfrag.x = (m < 8) ? b0 : 0.f;
        bfrag.y = (m < 8) ? b1 : 0.f;
        acc = __builtin_amdgcn_wmma_f32_16x16x4_f32(false, a, false, bfrag,
                                                    (short)0, acc, false, false);
    }

    // D layout: acc[g], lanes 0..15 -> C[g][t=lane]. Softmax over t (lanes 0..7).
    #pragma unroll
    for (int g = 0; g < GG; ++g) {
        const float v = acc[g] * 0.125f;   // 1/sqrt(d), d=64
        float mx = v;
        mx = fmaxf(mx, __shfl_xor(mx, 4, 32));
        mx = fmaxf(mx, __shfl_xor(mx, 2, 32));
        mx = fmaxf(mx, __shfl_xor(mx, 1, 32));
        const float e = __expf(v - mx);
        float s = e;
        s += __shfl_xor(s, 4, 32);
        s += __shfl_xor(s, 2, 32);
        s += __shfl_xor(s, 1, 32);
        if (lane < 8) attn[(b * GG + g) * TT + lane] = e / s;
    }
}

// ---------------- Kernel 3: fused t-weighted sum + depthwise 3x3 conv ----------
// out[b,c] = dwconv3x3( sum_t attn[b,g(c),t] * ctx[b,t,c] ), one block per (b,c).
__global__ void fuse_kernel(const float* __restrict__ ctx,
                            const float* __restrict__ wv,
                            const float* __restrict__ attn,
                            float* __restrict__ out)
{
    __shared__ float S[HWN];
    const int bc  = blockIdx.x;          // b*C + c
    const int b   = bc >> 9;
    const int c   = bc & (CC - 1);
    const int g   = c >> 6;
    const int tid = threadIdx.x;

    float a[TT];
    #pragma unroll
    for (int t = 0; t < TT; ++t) a[t] = attn[(b * GG + g) * TT + t];
    float w9[9];
    #pragma unroll
    for (int k = 0; k < 9; ++k) w9[k] = wv[c * 9 + k];

    // Phase 1: S = sum_t a[t] * ctx[b,t,c,:,:]   (float4 streaming)
    const size_t base = ((size_t)(b * TT) * CC + c) * HWN;
    for (int j = tid; j < HWN / 4; j += blockDim.x) {
        float4 s = make_float4(0.f, 0.f, 0.f, 0.f);
        #pragma unroll
        for (int t = 0; t < TT; ++t) {
            const float4 v = ((const float4*)(ctx + base + (size_t)t * CC * HWN))[j];
            s.x += a[t] * v.x; s.y += a[t] * v.y;
            s.z += a[t] * v.z; s.w += a[t] * v.w;
        }
        ((float4*)S)[j] = s;
    }
    __syncthreads();

    // Phase 2: 3x3 depthwise conv on S with zero padding
    for (int i = tid; i < HWN; i += blockDim.x) {
        const int h = i / WW;
        const int w = i - h * WW;
        float acc = 0.f;
        #pragma unroll
        for (int dy = -1; dy <= 1; ++dy) {
            const int hh = h + dy;
            #pragma unroll
            for (int dx = -1; dx <= 1; ++dx) {
                const int ww = w + dx;
                const float v = (hh >= 0 && hh < WW && ww >= 0 && ww < WW)
                                    ? S[hh * WW + ww] : 0.f;
                acc += v * w9[(dy + 1) * 3 + (dx + 1)];
            }
        }
        out[((size_t)b * CC + c) * HWN + i] = acc;
    }
}

extern "C" void kernel_launch(void* const* d_in, const int* in_sizes, int n_in,
                              void* d_out, int out_size, void* d_ws, size_t ws_size,
                              hipStream_t stream) {
    const float* x   = (const float*)d_in[0];
    const float* ctx = (const float*)d_in[1];
    const float* wq  = (const float*)d_in[2];
    const float* wk  = (const float*)d_in[3];
    const float* wv  = (const float*)d_in[4];
    float* out = (float*)d_out;
    float* ws  = (float*)d_ws;

    float* q_mean = ws;                       // 4096 floats
    float* k_mean = ws + BB * CC;             // 32768 floats
    float* attn   = ws + BB * CC + BB * TT * CC; // 512 floats

    const int rows = BB * CC + BB * TT * CC;  // 36864 reduction rows, 8 waves/block
    gap_kernel<<<rows / 8, 256, 0, stream>>>(x, ctx, q_mean, k_mean);
    attn_kernel<<<BB, 32, 0, stream>>>(q_mean, k_mean, wq, wk, attn);
    fuse_kernel<<<BB * CC, 256, 0, stream>>>(ctx, wv, attn, out);
}